// SinkhornOT_82471962018276
// MI455X (gfx1250) — compile-verified
//
#include <hip/hip_runtime.h>

#define D_MODEL 256
#define BATCH 16
#define NTOK 1024
#define NROWS (BATCH * NTOK)         // 16384
#define NN (NTOK * NTOK)             // 1048576
#define REG_INV 10.0f                // 1/REG
#define NEPS 1e-12f
#define NFBLK ((BATCH * NN) / 1024)  // 16384 finalize blocks

typedef __attribute__((ext_vector_type(2))) float v2f;
typedef __attribute__((ext_vector_type(4))) float v4f;
typedef __attribute__((ext_vector_type(8))) float v8f;

// ============================================================================
// Kernel 1: Y = l2_normalize(X @ W^T + bias), X:[NROWS,256], W:[256,256]
// grid.x = NROWS/16 row tiles, block = 128 threads (4 waves)
// ============================================================================
__global__ __launch_bounds__(128) void proj_norm_kernel(
    const float* __restrict__ X, const float* __restrict__ W,
    const float* __restrict__ bias, float* __restrict__ Y)
{
    const int S = 260;  // padded LDS row stride (16B-aligned rows, bank-spread)
    __shared__ float Xs[16 * 260];
    __shared__ float Ys[16 * 260];
    __shared__ float rn[16];

    const int tid  = threadIdx.x;
    const int wave = tid >> 5;
    const int lane = tid & 31;
    const int half = lane >> 4;
    const int lq   = lane & 15;
    const long rowbase = (long)blockIdx.x * 16;

    // Load X tile 16x256 (1024 float4s, 8 per thread)
    #pragma unroll
    for (int i = 0; i < 8; ++i) {
        int f4 = i * 128 + tid;
        int r  = f4 >> 6;         // 64 float4 per row
        int c4 = f4 & 63;
        v4f v = *(const v4f*)(X + (rowbase + r) * D_MODEL + c4 * 4);
        *(v4f*)(&Xs[r * S + c4 * 4]) = v;
    }
    __syncthreads();

    // Each wave computes 4 column tiles of 16
    for (int t = 0; t < 4; ++t) {
        const int jbase = (wave * 4 + t) * 16;
        v8f acc = {0.f, 0.f, 0.f, 0.f, 0.f, 0.f, 0.f, 0.f};
        const float* wrow = W + (long)(jbase + lq) * D_MODEL;
        const float* xrow = &Xs[lq * S];
        #pragma unroll 8
        for (int kk = 0; kk < D_MODEL; kk += 4) {
            const int ko = kk + 2 * half;
            v2f a  = *(const v2f*)(xrow + ko);
            v2f bb = *(const v2f*)(wrow + ko);
            acc = __builtin_amdgcn_wmma_f32_16x16x4_f32(
                false, a, false, bb, (short)0, acc, false, false);
        }
        const float bv = bias[jbase + lq];
        #pragma unroll
        for (int v = 0; v < 8; ++v)
            Ys[(v + 8 * half) * S + jbase + lq] = acc[v] + bv;
    }
    __syncthreads();

    // Row L2 norms
    if (tid < 16) {
        float s = 0.f;
        for (int c = 0; c < D_MODEL; ++c) { float y = Ys[tid * S + c]; s += y * y; }
        rn[tid] = 1.0f / fmaxf(sqrtf(s), NEPS);
    }
    __syncthreads();

    // Scale + store (coalesced)
    #pragma unroll
    for (int i = 0; i < 32; ++i) {
        int flat = i * 128 + tid;       // 0..4095
        int r = flat >> 8, c = flat & 255;
        Y[(rowbase + r) * D_MODEL + c] = Ys[r * S + c] * rn[r];
    }
}

// ============================================================================
// Kernel 2: cost[b,i,j] = 1 - q[b,i,:].k[b,j,:];  M = -cost/REG
// Block tile 32(i) x 64(j), 128 threads (4 waves). Each wave: one 16-row
// stripe x two 16-col tiles -> A-fragment reused across 2 WMMAs.
// grid (16 jtiles, 32 itiles, 16 batch)
// ============================================================================
__global__ __launch_bounds__(128) void cost_kernel(
    const float* __restrict__ Q, const float* __restrict__ K,
    float* __restrict__ cost, float* __restrict__ M)
{
    const int SC = 68;  // padded LDS stride
    __shared__ float Qs[32 * 68];
    __shared__ float Ks[64 * 68];

    const int tid  = threadIdx.x;
    const int wave = tid >> 5;
    const int lane = tid & 31;
    const int half = lane >> 4;
    const int lq   = lane & 15;
    const int jbase = blockIdx.x * 64;
    const int ibase = blockIdx.y * 32;
    const int b     = blockIdx.z;
    const float* Qb = Q + (long)b * NTOK * D_MODEL;
    const float* Kb = K + (long)b * NTOK * D_MODEL;
    const int sr  = (wave >> 1) * 16;   // subtile row offset within 32
    const int sc0 = (wave & 1) * 32;    // first col tile offset within 64

    v8f acc0 = {0.f, 0.f, 0.f, 0.f, 0.f, 0.f, 0.f, 0.f};
    v8f acc1 = {0.f, 0.f, 0.f, 0.f, 0.f, 0.f, 0.f, 0.f};

    for (int kc = 0; kc < 4; ++kc) {   // K chunks of 64
        // Load Q chunk 32x64 (512 float4, 4/thread)
        #pragma unroll
        for (int i = 0; i < 4; ++i) {
            int f4 = i * 128 + tid;
            int r  = f4 >> 4;          // 16 float4 per row
            int c4 = f4 & 15;
            *(v4f*)(&Qs[r * SC + c4 * 4]) =
                *(const v4f*)(Qb + (long)(ibase + r) * D_MODEL + kc * 64 + c4 * 4);
        }
        // Load K chunk 64x64 (1024 float4, 8/thread)
        #pragma unroll
        for (int i = 0; i < 8; ++i) {
            int f4 = i * 128 + tid;
            int r  = f4 >> 4;
            int c4 = f4 & 15;
            *(v4f*)(&Ks[r * SC + c4 * 4]) =
                *(const v4f*)(Kb + (long)(jbase + r) * D_MODEL + kc * 64 + c4 * 4);
        }
        __syncthreads();
        const float* qrow = &Qs[(sr + lq) * SC];
        const float* k0   = &Ks[(sc0 + lq) * SC];
        const float* k1   = &Ks[(sc0 + 16 + lq) * SC];
        #pragma unroll 8
        for (int kk = 0; kk < 64; kk += 4) {
            const int ko = kk + 2 * half;
            v2f a  = *(const v2f*)(qrow + ko);
            v2f b0 = *(const v2f*)(k0 + ko);
            v2f b1 = *(const v2f*)(k1 + ko);
            acc0 = __builtin_amdgcn_wmma_f32_16x16x4_f32(
                false, a, false, b0, (short)0, acc0, false, false);
            acc1 = __builtin_amdgcn_wmma_f32_16x16x4_f32(
                false, a, false, b1, (short)0, acc1, false, false);
        }
        __syncthreads();
    }

    #pragma unroll
    for (int v = 0; v < 8; ++v) {
        const int gi  = ibase + sr + v + 8 * half;
        const int gj0 = jbase + sc0 + lq;
        const long idx0 = (long)b * NN + (long)gi * NTOK + gj0;
        const float c0 = 1.0f - acc0[v];
        const float c1 = 1.0f - acc1[v];
        cost[idx0]      = c0;
        cost[idx0 + 16] = c1;
        M[idx0]         = -c0 * REG_INV;
        M[idx0 + 16]    = -c1 * REG_INV;
    }
}

// ============================================================================
// Kernel 3: row logsumexp subtract (axis=2). One block per (b,i) row.
// ============================================================================
__global__ __launch_bounds__(256) void row_lse_kernel(float* __restrict__ M)
{
    __shared__ float red[256];
    float* Mr = M + (long)blockIdx.x * NTOK;
    const int t = threadIdx.x;
    float x0 = Mr[t], x1 = Mr[t + 256], x2 = Mr[t + 512], x3 = Mr[t + 768];
    float mx = fmaxf(fmaxf(x0, x1), fmaxf(x2, x3));
    red[t] = mx; __syncthreads();
    for (int s = 128; s > 0; s >>= 1) {
        if (t < s) red[t] = fmaxf(red[t], red[t + s]);
        __syncthreads();
    }
    mx = red[0]; __syncthreads();
    float sum = expf(x0 - mx) + expf(x1 - mx) + expf(x2 - mx) + expf(x3 - mx);
    red[t] = sum; __syncthreads();
    for (int s = 128; s > 0; s >>= 1) {
        if (t < s) red[t] += red[t + s];
        __syncthreads();
    }
    const float lse = mx + logf(red[0]);
    Mr[t]       = x0 - lse;
    Mr[t + 256] = x1 - lse;
    Mr[t + 512] = x2 - lse;
    Mr[t + 768] = x3 - lse;
}

// ============================================================================
// Kernel 4: column logsumexp subtract (axis=1). Thread owns one column j;
// consecutive threads -> consecutive j -> coalesced strided sweeps.
// grid (4 col-chunks, 16 batch), block 256.
// ============================================================================
__global__ __launch_bounds__(256) void col_lse_kernel(float* __restrict__ M)
{
    const int j = blockIdx.x * 256 + threadIdx.x;
    float* Mb = M + (long)blockIdx.y * NN;
    float mx = -INFINITY;
    for (int i = 0; i < NTOK; ++i) mx = fmaxf(mx, Mb[(long)i * NTOK + j]);
    float sum = 0.f;
    for (int i = 0; i < NTOK; ++i) sum += expf(Mb[(long)i * NTOK + j] - mx);
    const float lse = mx + logf(sum);
    for (int i = 0; i < NTOK; ++i) Mb[(long)i * NTOK + j] -= lse;
}

// ============================================================================
// Kernel 5: plan = exp(M) in place; per-block partial sums of plan*cost
// written to bsum[] (deterministic, no float atomics).
// ============================================================================
__global__ __launch_bounds__(256) void finalize_kernel(
    float* __restrict__ MP, const float* __restrict__ cost,
    float* __restrict__ bsum)
{
    __shared__ float red[256];
    const long base = (long)blockIdx.x * 1024;
    const int t = threadIdx.x;
    float s = 0.f;
    #pragma unroll
    for (int e = 0; e < 4; ++e) {
        const long idx = base + e * 256 + t;
        const float p = expf(MP[idx]);
        MP[idx] = p;
        s += p * cost[idx];
    }
    red[t] = s; __syncthreads();
    for (int st = 128; st > 0; st >>= 1) {
        if (t < st) red[t] += red[t + st];
        __syncthreads();
    }
    if (t == 0) bsum[blockIdx.x] = red[0];
}

// ============================================================================
// Kernel 6: deterministic final reduction of NFBLK partial sums -> ot_cost
// Single block of 256; fixed summation order on every call.
// ============================================================================
__global__ __launch_bounds__(256) void ot_reduce_kernel(
    const float* __restrict__ bsum, float* __restrict__ ot)
{
    __shared__ float red[256];
    const int t = threadIdx.x;
    float s = 0.f;
    for (int i = t; i < NFBLK; i += 256) s += bsum[i];
    red[t] = s; __syncthreads();
    for (int st = 128; st > 0; st >>= 1) {
        if (t < st) red[t] += red[t + st];
        __syncthreads();
    }
    if (t == 0) ot[0] = red[0] * (1.0f / BATCH);
}

// ============================================================================
extern "C" void kernel_launch(void* const* d_in, const int* in_sizes, int n_in,
                              void* d_out, int out_size, void* d_ws, size_t ws_size,
                              hipStream_t stream)
{
    const float* drone = (const float*)d_in[0];
    const float* sat   = (const float*)d_in[1];
    const float* Wq    = (const float*)d_in[2];
    const float* bq    = (const float*)d_in[3];
    const float* Wk    = (const float*)d_in[4];
    const float* bk    = (const float*)d_in[5];

    float* out  = (float*)d_out;
    float* Qn   = (float*)d_ws;                       // 16 MB
    float* Kn   = Qn + (size_t)NROWS * D_MODEL;       // 16 MB
    float* cost = Kn + (size_t)NROWS * D_MODEL;       // 64 MB
    float* bsum = cost + (size_t)BATCH * NN;          // 64 KB
    float* M    = out;                                // Sinkhorn buffer lives in d_out plan slot
    float* ot   = out + (size_t)BATCH * NN;           // scalar at end

    proj_norm_kernel<<<NROWS / 16, 128, 0, stream>>>(drone, Wq, bq, Qn);
    proj_norm_kernel<<<NROWS / 16, 128, 0, stream>>>(sat,   Wk, bk, Kn);

    cost_kernel<<<dim3(16, 32, BATCH), 128, 0, stream>>>(Qn, Kn, cost, M);

    for (int it = 0; it < 5; ++it) {
        row_lse_kernel<<<NROWS, 256, 0, stream>>>(M);
        col_lse_kernel<<<dim3(4, BATCH), 256, 0, stream>>>(M);
    }

    finalize_kernel<<<NFBLK, 256, 0, stream>>>(M, cost, bsum);
    ot_reduce_kernel<<<1, 256, 0, stream>>>(bsum, ot);
}